// SAENet_88227218195257
// MI455X (gfx1250) — compile-verified
//
#include <hip/hip_runtime.h>
#include <hip/hip_bf16.h>

// ---------------------------------------------------------------------------
// SAE forward for MI455X (gfx1250, wave32, WMMA + async LDS DMA).
//   beff   = b1 - W1 @ b2                         (tiny precompute)
//   f      = relu(x @ W1^T + beff)                [B,S]  f32 WMMA 16x16x4,
//            both tiles via GLOBAL_LOAD_ASYNC_TO_LDS_B128, double-buffered
//   batch top-K via IEEE-bit histogram threshold (fused into encoder epilogue)
//   x_rec  = f_sparse @ W2^T + b2                 [B,H]  WMMA, masked A tiles
//   dead_x = dead_sparse @ W2^T + b2              same pass, 2nd accumulator
//   f_sparse written in-place into d_out's middle region.
// ---------------------------------------------------------------------------

#define B_SZ      4096
#define H_SZ      512
#define S_SZ      16384
#define K_TOP     32
#define DEAD_K    16
#define DEAD_THR  5

#define NBINS     2304          // bins of (bits >> 19): covers (0, 131072.0)
#define BIN_SHIFT 19
#define TILE_K    16
#define LDT       20            // padded LDS stride (floats); 80B = 16B-aligned

typedef float v2f __attribute__((ext_vector_type(2)));
typedef float v8f __attribute__((ext_vector_type(8)));
typedef int   v4i __attribute__((ext_vector_type(4)));

// ---------------------------------------------------- async copy primitives
#if defined(__gfx1250__) && __has_builtin(__builtin_amdgcn_global_load_async_to_lds_b128)
#define SAE_ASYNC 1
#endif

__device__ __forceinline__ void sae_copy16(const float* g, float* l) {
#if SAE_ASYNC
    __builtin_amdgcn_global_load_async_to_lds_b128(
        (__attribute__((address_space(1))) v4i*)g,
        (__attribute__((address_space(3))) v4i*)l, 0, 0);
#else
    *(float4*)l = *(const float4*)g;
#endif
}

__device__ __forceinline__ void sae_wait_async() {
#if SAE_ASYNC
#if __has_builtin(__builtin_amdgcn_s_wait_asynccnt)
    __builtin_amdgcn_s_wait_asynccnt(0);
#else
    asm volatile("s_wait_asynccnt 0x0" ::: "memory");
#endif
#endif
}

// -------------------------------------------------------------------- utils
__global__ void sae_zero_ws(unsigned* __restrict__ p, int n) {
    int i = blockIdx.x * blockDim.x + threadIdx.x;
    if (i < n) p[i] = 0u;
}

// beff[n] = b1[n] - sum_k b2[k] * W1[n,k]   (folds encoder pre-bias out of
// the hot loop so the A-tile becomes a raw copy -> async DMA eligible)
__global__ __launch_bounds__(256)
void sae_enc_bias(const float* __restrict__ W1, const float* __restrict__ b1,
                  const float* __restrict__ b2, float* __restrict__ beff) {
    __shared__ float b2s[H_SZ];
    for (int i = threadIdx.x; i < H_SZ; i += 256) b2s[i] = b2[i];
    __syncthreads();
    const int n = blockIdx.x * 256 + threadIdx.x;
    float s = 0.0f;
    for (int k = 0; k < H_SZ; k += 4) {
        const float4 w = *(const float4*)(W1 + (size_t)n * H_SZ + k);
        s += w.x * b2s[k] + w.y * b2s[k + 1] + w.z * b2s[k + 2] + w.w * b2s[k + 3];
    }
    beff[n] = b1[n] - s;
}

// ---------------------------------------------------------------- encoder
// grid = (S/128, B/128), block = 256 (8 waves). Wave tile 64x32 = 4x2 WMMAs.
// Double-buffered LDS tiles fed by async b128 DMA.
__global__ __launch_bounds__(256)
void sae_encoder(const float* __restrict__ x, const float* __restrict__ W1,
                 const float* __restrict__ beff,
                 const int* __restrict__ dead_features,
                 float* __restrict__ f_out,
                 unsigned* __restrict__ hist_live,
                 unsigned* __restrict__ hist_dead) {
    __shared__ float As[2][128 * LDT];       // x tile        [row][k]
    __shared__ float Bs[2][128 * LDT];       // W1 tile       [n][k]
    __shared__ unsigned h_live[NBINS];
    __shared__ unsigned h_dead[NBINS];

    const int tid  = threadIdx.x;
    const int lane = tid & 31;
    const int wave = tid >> 5;
    const int bm   = blockIdx.y * 128;       // batch rows
    const int bn   = blockIdx.x * 128;       // codebook cols

    for (int i = tid; i < NBINS; i += 256) { h_live[i] = 0u; h_dead[i] = 0u; }

    const int wm   = (wave & 1) * 64;
    const int wn   = (wave >> 1) * 32;
    const int lrow = lane & 15;
    const int kb   = (lane >> 4) * 2;        // 16x16x4 f32 A/B frag k-base

    v8f acc[4][2];
    #pragma unroll
    for (int i = 0; i < 4; ++i)
        #pragma unroll
        for (int j = 0; j < 2; ++j)
            #pragma unroll
            for (int r = 0; r < 8; ++r) acc[i][j][r] = 0.0f;

    const int row0 = tid >> 2;               // thread's tile-copy coordinates
    const int row1 = (tid + 256) >> 2;
    const int c40  = (tid & 3) * 4;

    // issue tile 0
    sae_copy16(x  + (size_t)(bm + row0) * H_SZ + c40, &As[0][row0 * LDT + c40]);
    sae_copy16(x  + (size_t)(bm + row1) * H_SZ + c40, &As[0][row1 * LDT + c40]);
    sae_copy16(W1 + (size_t)(bn + row0) * H_SZ + c40, &Bs[0][row0 * LDT + c40]);
    sae_copy16(W1 + (size_t)(bn + row1) * H_SZ + c40, &Bs[0][row1 * LDT + c40]);

    const int NT = H_SZ / TILE_K;
    for (int t = 0; t < NT; ++t) {
        sae_wait_async();
        __syncthreads();
        if (t + 1 < NT) {                    // issue tile t+1 into other buffer
            const int kn = (t + 1) * TILE_K;
            float* an = As[(t + 1) & 1];
            float* bnp = Bs[(t + 1) & 1];
            sae_copy16(x  + (size_t)(bm + row0) * H_SZ + kn + c40, an  + row0 * LDT + c40);
            sae_copy16(x  + (size_t)(bm + row1) * H_SZ + kn + c40, an  + row1 * LDT + c40);
            sae_copy16(W1 + (size_t)(bn + row0) * H_SZ + kn + c40, bnp + row0 * LDT + c40);
            sae_copy16(W1 + (size_t)(bn + row1) * H_SZ + kn + c40, bnp + row1 * LDT + c40);
        }
        const float* Ab = As[t & 1];
        const float* Bb = Bs[t & 1];
        #pragma unroll
        for (int kk = 0; kk < TILE_K; kk += 4) {
            v2f af[4], bf[2];
            #pragma unroll
            for (int i = 0; i < 4; ++i) {
                const int m = wm + i * 16 + lrow;
                af[i].x = Ab[m * LDT + kk + kb];
                af[i].y = Ab[m * LDT + kk + kb + 1];
            }
            #pragma unroll
            for (int j = 0; j < 2; ++j) {
                const int n = wn + j * 16 + lrow;
                bf[j].x = Bb[n * LDT + kk + kb];
                bf[j].y = Bb[n * LDT + kk + kb + 1];
            }
            #pragma unroll
            for (int i = 0; i < 4; ++i)
                #pragma unroll
                for (int j = 0; j < 2; ++j)
                    acc[i][j] = __builtin_amdgcn_wmma_f32_16x16x4_f32(
                        false, af[i], false, bf[j], (short)0, acc[i][j],
                        false, false);
        }
    }

    // epilogue: + beff, relu, store f, fused histogram of positives
    #pragma unroll
    for (int j = 0; j < 2; ++j) {
        const int n = bn + wn + j * 16 + lrow;
        const float bias   = beff[n];
        const bool is_dead = (dead_features[n] >= DEAD_THR);
        #pragma unroll
        for (int i = 0; i < 4; ++i) {
            #pragma unroll
            for (int r = 0; r < 8; ++r) {
                const int m = bm + wm + i * 16 + r + 8 * (lane >> 4);
                float v = acc[i][j][r] + bias;
                v = v > 0.0f ? v : 0.0f;
                f_out[(size_t)m * S_SZ + n] = v;
                if (v > 0.0f) {
                    unsigned bin = __float_as_uint(v) >> BIN_SHIFT;
                    if (bin >= NBINS) bin = NBINS - 1;
                    atomicAdd(&h_live[bin], 1u);
                    if (is_dead) atomicAdd(&h_dead[bin], 1u);
                }
            }
        }
    }
    __syncthreads();
    for (int i = tid; i < NBINS; i += 256) {
        unsigned c = h_live[i]; if (c) atomicAdd(&hist_live[i], c);
        unsigned d = h_dead[i]; if (d) atomicAdd(&hist_dead[i], d);
    }
}

// -------------------------------------------------------------- threshold
__global__ void sae_threshold(const unsigned* __restrict__ hist_live,
                              const unsigned* __restrict__ hist_dead,
                              unsigned* __restrict__ thresh) {
    if (threadIdx.x == 0 && blockIdx.x == 0) {
        long long cum = 0; unsigned tb = 0u;
        for (int b = NBINS - 1; b >= 0; --b) {
            cum += hist_live[b];
            if (cum >= (long long)K_TOP * B_SZ) { tb = (unsigned)b << BIN_SHIFT; break; }
        }
        thresh[0] = tb;
        cum = 0; tb = 0u;
        for (int b = NBINS - 1; b >= 0; --b) {
            cum += hist_dead[b];
            if (cum >= (long long)DEAD_K * B_SZ) { tb = (unsigned)b << BIN_SHIFT; break; }
        }
        thresh[1] = tb;
    }
}

// ---------------------------------------------------------------- decoder
// grid = (H/128, B/128), block = 256. Dual accumulators (live / dead) share
// the W2 B-fragments. W2 tile arrives via async DMA while the VALU masks the
// f tile (thresholded view == f_sparse / dead_sparse) into LDS.
__global__ __launch_bounds__(256)
void sae_decode(const float* __restrict__ f, const float* __restrict__ W2,
                const float* __restrict__ b2, const int* __restrict__ dead_features,
                const unsigned* __restrict__ thresh,
                float* __restrict__ x_rec, float* __restrict__ dead_x) {
    __shared__ float Al[128 * LDT];          // live-masked f tile [row][k]
    __shared__ float Ad[128 * LDT];          // dead-masked f tile
    __shared__ float Bs[128 * LDT];          // W2 tile [h][k=s]

    const int tid  = threadIdx.x;
    const int lane = tid & 31;
    const int wave = tid >> 5;
    const int bm   = blockIdx.y * 128;       // batch rows
    const int bn   = blockIdx.x * 128;       // hidden cols

    const unsigned tl = thresh[0];
    const unsigned td = thresh[1];

    const int wm   = (wave & 1) * 64;
    const int wn   = (wave >> 1) * 32;
    const int lrow = lane & 15;
    const int kb   = (lane >> 4) * 2;

    v8f accl[4][2], accd[4][2];
    #pragma unroll
    for (int i = 0; i < 4; ++i)
        #pragma unroll
        for (int j = 0; j < 2; ++j)
            #pragma unroll
            for (int r = 0; r < 8; ++r) { accl[i][j][r] = 0.0f; accd[i][j][r] = 0.0f; }

    const int row0 = tid >> 2;
    const int row1 = (tid + 256) >> 2;
    const int c40  = (tid & 3) * 4;

    for (int k0 = 0; k0 < S_SZ; k0 += TILE_K) {
        // async DMA of the raw W2 tile (overlaps the masking work below)
        sae_copy16(W2 + (size_t)(bn + row0) * S_SZ + k0 + c40, &Bs[row0 * LDT + c40]);
        sae_copy16(W2 + (size_t)(bn + row1) * S_SZ + k0 + c40, &Bs[row1 * LDT + c40]);

        // threshold-masked f tiles through the VALU
        #pragma unroll
        for (int it = 0; it < 2; ++it) {
            const int row = it ? row1 : row0;
            const float* src = f + (size_t)(bm + row) * S_SZ + k0 + c40;
            if (k0 + TILE_K < S_SZ) __builtin_prefetch(src + TILE_K, 0, 0);
            const float4 fv = *(const float4*)src;
            const float fe[4] = {fv.x, fv.y, fv.z, fv.w};
            #pragma unroll
            for (int e = 0; e < 4; ++e) {
                const float    val  = fe[e];
                const unsigned bits = __float_as_uint(val);
                const bool     pos  = val > 0.0f;
                const bool     dm   = dead_features[k0 + c40 + e] >= DEAD_THR;
                Al[row * LDT + c40 + e] = (pos && bits >= tl) ? val : 0.0f;
                Ad[row * LDT + c40 + e] = (pos && dm && bits >= td) ? val : 0.0f;
            }
        }
        sae_wait_async();
        __syncthreads();

        #pragma unroll
        for (int kk = 0; kk < TILE_K; kk += 4) {
            v2f afl[4], afd[4], bf[2];
            #pragma unroll
            for (int i = 0; i < 4; ++i) {
                const int m = wm + i * 16 + lrow;
                afl[i].x = Al[m * LDT + kk + kb];
                afl[i].y = Al[m * LDT + kk + kb + 1];
                afd[i].x = Ad[m * LDT + kk + kb];
                afd[i].y = Ad[m * LDT + kk + kb + 1];
            }
            #pragma unroll
            for (int j = 0; j < 2; ++j) {
                const int n = wn + j * 16 + lrow;
                bf[j].x = Bs[n * LDT + kk + kb];
                bf[j].y = Bs[n * LDT + kk + kb + 1];
            }
            #pragma unroll
            for (int i = 0; i < 4; ++i)
                #pragma unroll
                for (int j = 0; j < 2; ++j) {
                    accl[i][j] = __builtin_amdgcn_wmma_f32_16x16x4_f32(
                        false, afl[i], false, bf[j], (short)0, accl[i][j], false, false);
                    accd[i][j] = __builtin_amdgcn_wmma_f32_16x16x4_f32(
                        false, afd[i], false, bf[j], (short)0, accd[i][j], false, false);
                }
        }
        __syncthreads();
    }

    #pragma unroll
    for (int j = 0; j < 2; ++j) {
        const int n = bn + wn + j * 16 + lrow;
        const float bias = b2[n];
        #pragma unroll
        for (int i = 0; i < 4; ++i) {
            #pragma unroll
            for (int r = 0; r < 8; ++r) {
                const int m = bm + wm + i * 16 + r + 8 * (lane >> 4);
                x_rec [(size_t)m * H_SZ + n] = accl[i][j][r] + bias;
                dead_x[(size_t)m * H_SZ + n] = accd[i][j][r] + bias;
            }
        }
    }
}

// ---------------------------------------------------------------- sparsify
__global__ __launch_bounds__(256)
void sae_sparsify(float* __restrict__ fio, const unsigned* __restrict__ thresh) {
    const unsigned tl = thresh[0];
    const size_t vi = (size_t)blockIdx.x * blockDim.x + threadIdx.x;
    float4* p = (float4*)fio + vi;
    float4 v = *p;
    float fe[4] = {v.x, v.y, v.z, v.w};
    #pragma unroll
    for (int e = 0; e < 4; ++e) {
        const bool keep = (fe[e] > 0.0f) && (__float_as_uint(fe[e]) >= tl);
        fe[e] = keep ? fe[e] : 0.0f;
    }
    v.x = fe[0]; v.y = fe[1]; v.z = fe[2]; v.w = fe[3];
    *p = v;
}

// ------------------------------------------------------------------- launch
extern "C" void kernel_launch(void* const* d_in, const int* in_sizes, int n_in,
                              void* d_out, int out_size, void* d_ws, size_t ws_size,
                              hipStream_t stream) {
    const float* x    = (const float*)d_in[0];
    const float* W1   = (const float*)d_in[1];
    const float* b1   = (const float*)d_in[2];
    const float* W2   = (const float*)d_in[3];
    const float* b2   = (const float*)d_in[4];
    const int*   dead = (const int*)  d_in[5];

    float* out    = (float*)d_out;
    float* x_rec  = out;                                   // [B,H]
    float* f_reg  = out + (size_t)B_SZ * H_SZ;             // [B,S] (raw f -> f_sparse)
    float* dead_x = out + (size_t)B_SZ * H_SZ + (size_t)B_SZ * S_SZ;  // [B,H]

    unsigned* ws        = (unsigned*)d_ws;
    unsigned* hist_live = ws;                              // [NBINS]
    unsigned* hist_dead = ws + NBINS;                      // [NBINS]
    unsigned* thresh    = ws + 2 * NBINS;                  // [2]
    float*    beff      = (float*)(ws + 2 * NBINS + 2);    // [S]

    sae_zero_ws<<<dim3((2 * NBINS + 255) / 256), dim3(256), 0, stream>>>(ws, 2 * NBINS);

    sae_enc_bias<<<dim3(S_SZ / 256), dim3(256), 0, stream>>>(W1, b1, b2, beff);

    sae_encoder<<<dim3(S_SZ / 128, B_SZ / 128), dim3(256), 0, stream>>>(
        x, W1, beff, dead, f_reg, hist_live, hist_dead);

    sae_threshold<<<dim3(1), dim3(32), 0, stream>>>(hist_live, hist_dead, thresh);

    sae_decode<<<dim3(H_SZ / 128, B_SZ / 128), dim3(256), 0, stream>>>(
        f_reg, W2, b2, dead, thresh, x_rec, dead_x);

    sae_sparsify<<<dim3(((size_t)B_SZ * S_SZ / 4 + 255) / 256), dim3(256), 0, stream>>>(
        f_reg, thresh);
}